// MHSS_SSSM_Block_35038343201495
// MI455X (gfx1250) — compile-verified
//
#include <hip/hip_runtime.h>
#include <hip/hip_bf16.h>
#include <math.h>

// ---------------- problem constants ----------------
#define DMODEL 256
#define LSEQ   1024
#define BATCH  4
#define NDIR   4
#define NST    16     // d_state
#define RDT    16     // dt_rank
#define HH     32
#define WW     32
#define XPO    48     // R + 2N

typedef __attribute__((ext_vector_type(2))) float v2f;
typedef __attribute__((ext_vector_type(8))) float v8f;

__device__ __forceinline__ float silu_f(float v) {
  return v / (1.0f + __expf(-v));
}

#define WMMA_F32(A, B, C) \
  __builtin_amdgcn_wmma_f32_16x16x4_f32(false, (A), false, (B), (short)0, (C), false, false)

// ---------------------------------------------------------------------------
// Kernel 1: xz = X_dir @ in_w^T   (per direction, gathered rows)
// Register-blocked: 4 N-tiles (64 output cols) per wave.
// grid = (8, NDIR*BATCH*LSEQ/16), block = 1 wave (32)
// ---------------------------------------------------------------------------
__global__ __launch_bounds__(32) void k_inproj(const float* __restrict__ x,
                                               const float* __restrict__ in_w,
                                               float* __restrict__ xz) {
  const int ntg   = blockIdx.x;                // 0..7, 64 cols each
  const int mtile = blockIdx.y;                // 0..1023
  const int dir = mtile >> 8;                  // 256 M-tiles per direction
  const int rem = mtile & 255;
  const int b   = rem >> 6;                    // 64 l-tiles per batch
  const int l0  = (rem & 63) << 4;
  const int t   = threadIdx.x;
  const int mrow  = t & 15;
  const int khalf = (t >> 4) << 1;             // 0 or 2
  const int col   = t & 15;
  const int hi    = t >> 4;
  const int n0    = ntg << 6;

  // direction-local row -> flat spatial position in (H,W)
  const int l = l0 + mrow;
  int pos;
  if (dir == 0)      pos = l;
  else if (dir == 1) pos = (LSEQ - 1) - l;
  else {
    const int lv = (dir == 2) ? l : (LSEQ - 1) - l;
    pos = (lv & 31) * WW + (lv >> 5);          // x_v[l] = x_flat[h*W + w], l = w*H + h
  }

  const float* xb = x + (size_t)b * DMODEL * LSEQ;                        // (C,H*W)
  const float* w0 = in_w + ((size_t)dir * 2 * DMODEL + n0 + col) * DMODEL;
  const float* w1 = w0 + (size_t)16 * DMODEL;
  const float* w2 = w0 + (size_t)32 * DMODEL;
  const float* w3 = w0 + (size_t)48 * DMODEL;

  v8f a0 = {}, a1 = {}, a2 = {}, a3 = {};
  for (int k0 = 0; k0 < DMODEL; k0 += 4) {
    const int ka = k0 + khalf;
    v2f a;  a.x = xb[(size_t)ka * LSEQ + pos];
            a.y = xb[(size_t)(ka + 1) * LSEQ + pos];
    v2f b0; b0.x = w0[ka]; b0.y = w0[ka + 1];
    v2f b1; b1.x = w1[ka]; b1.y = w1[ka + 1];
    v2f b2; b2.x = w2[ka]; b2.y = w2[ka + 1];
    v2f b3; b3.x = w3[ka]; b3.y = w3[ka + 1];
    a0 = WMMA_F32(a, b0, a0);
    a1 = WMMA_F32(a, b1, a1);
    a2 = WMMA_F32(a, b2, a2);
    a3 = WMMA_F32(a, b3, a3);
  }

  float* o = xz + (((size_t)(dir * BATCH + b) * LSEQ + l0) * (2 * DMODEL))
                + n0 + col;
#pragma unroll
  for (int r = 0; r < 8; ++r) {
    float* orow = o + (size_t)(hi * 8 + r) * (2 * DMODEL);
    orow[0]  = a0[r];
    orow[16] = a1[r];
    orow[32] = a2[r];
    orow[48] = a3[r];
  }
}

// ---------------------------------------------------------------------------
// Kernel 2: depthwise causal conv(4) + bias + SiLU  -> xc
// ---------------------------------------------------------------------------
__global__ __launch_bounds__(256) void k_conv(const float* __restrict__ xz,
                                              const float* __restrict__ cw,
                                              const float* __restrict__ cb,
                                              float* __restrict__ xc) {
  const size_t i = (size_t)blockIdx.x * blockDim.x + threadIdx.x;
  const size_t total = (size_t)NDIR * BATCH * LSEQ * DMODEL;
  if (i >= total) return;
  const int c  = (int)(i & (DMODEL - 1));
  const size_t lb = i >> 8;                    // (dir*B + b)*L + l
  const int l  = (int)(lb & (LSEQ - 1));
  const int dir = (int)(lb >> 12);             // /(B*L)

  const float* w = cw + ((size_t)dir * DMODEL + c) * 4;
  float acc = cb[dir * DMODEL + c];
#pragma unroll
  for (int k = 0; k < 4; ++k) {
    const int ls = l + k - 3;
    if (ls >= 0)
      acc += xz[((lb + (size_t)(k - 3)) << 9) + c] * w[k];
  }
  xc[i] = silu_f(acc);
}

// ---------------------------------------------------------------------------
// Kernel 3: dbl = xc @ xp_w^T   (48 outputs: dt_r | B | C) — all 3 N-tiles per wave
// grid = (1024), block = 1 wave
// ---------------------------------------------------------------------------
__global__ __launch_bounds__(32) void k_xproj(const float* __restrict__ xc,
                                              const float* __restrict__ xp_w,
                                              float* __restrict__ dbl) {
  const int mtile = blockIdx.x;                // 0..1023
  const int dir = mtile >> 8;
  const int t = threadIdx.x;
  const int mrow = t & 15, khalf = (t >> 4) << 1, col = t & 15, hi = t >> 4;
  const size_t rowbase = (size_t)mtile * 16;

  const float* w0 = xp_w + ((size_t)dir * XPO + col) * DMODEL;
  const float* w1 = w0 + (size_t)16 * DMODEL;
  const float* w2 = w0 + (size_t)32 * DMODEL;

  v8f a0 = {}, a1 = {}, a2 = {};
  for (int k0 = 0; k0 < DMODEL; k0 += 4) {
    const int ka = k0 + khalf;
    const float* ar = xc + (rowbase + mrow) * DMODEL + ka;
    v2f a;  a.x = ar[0]; a.y = ar[1];
    v2f b0; b0.x = w0[ka]; b0.y = w0[ka + 1];
    v2f b1; b1.x = w1[ka]; b1.y = w1[ka + 1];
    v2f b2; b2.x = w2[ka]; b2.y = w2[ka + 1];
    a0 = WMMA_F32(a, b0, a0);
    a1 = WMMA_F32(a, b1, a1);
    a2 = WMMA_F32(a, b2, a2);
  }
#pragma unroll
  for (int r = 0; r < 8; ++r) {
    float* orow = dbl + (rowbase + hi * 8 + r) * XPO + col;
    orow[0]  = a0[r];
    orow[16] = a1[r];
    orow[32] = a2[r];
  }
}

// ---------------------------------------------------------------------------
// Kernel 4: dt = softplus(dt_r @ dt_w^T + dt_b)   (K = 16, 4 N-tiles per wave)
// grid = (4, 1024), block = 1 wave
// ---------------------------------------------------------------------------
__global__ __launch_bounds__(32) void k_dtproj(const float* __restrict__ dbl,
                                               const float* __restrict__ dt_w,
                                               const float* __restrict__ dt_b,
                                               float* __restrict__ dt) {
  const int ntg   = blockIdx.x;                // 0..3, 64 cols each
  const int mtile = blockIdx.y;                // 0..1023
  const int dir = mtile >> 8;
  const int t = threadIdx.x;
  const int mrow = t & 15, khalf = (t >> 4) << 1, col = t & 15, hi = t >> 4;
  const size_t rowbase = (size_t)mtile * 16;
  const int n0 = ntg << 6;

  const float* w0 = dt_w + ((size_t)dir * DMODEL + n0 + col) * RDT;
  const float* w1 = w0 + 16 * RDT;
  const float* w2 = w0 + 32 * RDT;
  const float* w3 = w0 + 48 * RDT;

  v8f a0 = {}, a1 = {}, a2 = {}, a3 = {};
#pragma unroll
  for (int k0 = 0; k0 < RDT; k0 += 4) {
    const int ka = k0 + khalf;
    const float* ar = dbl + (rowbase + mrow) * XPO + ka;   // dt_r = dbl[:, 0:16]
    v2f a;  a.x = ar[0]; a.y = ar[1];
    v2f b0; b0.x = w0[ka]; b0.y = w0[ka + 1];
    v2f b1; b1.x = w1[ka]; b1.y = w1[ka + 1];
    v2f b2; b2.x = w2[ka]; b2.y = w2[ka + 1];
    v2f b3; b3.x = w3[ka]; b3.y = w3[ka + 1];
    a0 = WMMA_F32(a, b0, a0);
    a1 = WMMA_F32(a, b1, a1);
    a2 = WMMA_F32(a, b2, a2);
    a3 = WMMA_F32(a, b3, a3);
  }
  const float* bb = dt_b + dir * DMODEL + n0 + col;
  const float bias0 = bb[0], bias1 = bb[16], bias2 = bb[32], bias3 = bb[48];
#pragma unroll
  for (int r = 0; r < 8; ++r) {
    float* orow = dt + (rowbase + hi * 8 + r) * DMODEL + n0 + col;
    float v0 = a0[r] + bias0; v0 = (v0 > 20.0f) ? v0 : log1pf(__expf(v0));
    float v1 = a1[r] + bias1; v1 = (v1 > 20.0f) ? v1 : log1pf(__expf(v1));
    float v2 = a2[r] + bias2; v2 = (v2 > 20.0f) ? v2 : log1pf(__expf(v2));
    float v3 = a3[r] + bias3; v3 = (v3 > 20.0f) ? v3 : log1pf(__expf(v3));
    orow[0] = v0; orow[16] = v1; orow[32] = v2; orow[48] = v3;
  }
}

// ---------------------------------------------------------------------------
// Kernel 5: selective scan, fused D-skip + SiLU(z) gate.
// One block per (dir,b); thread = channel; h[16] in registers; B/C in LDS.
// Writes yz in place over xc.
// ---------------------------------------------------------------------------
__global__ __launch_bounds__(256) void k_scan(const float* __restrict__ dbl,
                                              const float* __restrict__ dt,
                                              const float* __restrict__ xz,
                                              const float* __restrict__ A_log,
                                              const float* __restrict__ Dp,
                                              float* __restrict__ xc) {
  __shared__ float sBC[64 * 32];               // 64 steps x (B[16] | C[16])
  const int db  = blockIdx.x;                  // dir*B + b
  const int dir = db >> 2;
  const int c   = threadIdx.x;

  float Arow[NST];
#pragma unroll
  for (int n = 0; n < NST; ++n)
    Arow[n] = -__expf(A_log[((size_t)dir * DMODEL + c) * NST + n]);
  const float dskip = Dp[dir * DMODEL + c];

  float h[NST];
#pragma unroll
  for (int n = 0; n < NST; ++n) h[n] = 0.0f;

  const size_t base = (size_t)db * LSEQ;
  for (int chunk = 0; chunk < LSEQ / 64; ++chunk) {
    const int lc0 = chunk * 64;
    for (int idx = threadIdx.x; idx < 64 * 32; idx += 256) {
      const int lp = idx >> 5, j = idx & 31;
      sBC[idx] = dbl[(base + lc0 + lp) * XPO + RDT + j];
    }
    __syncthreads();
    for (int lp = 0; lp < 64; ++lp) {
      const size_t row = base + lc0 + lp;
      const float dtv = dt[row * DMODEL + c];
      const float xcv = xc[row * DMODEL + c];
      const float u = dtv * xcv;
      const float* Bm = &sBC[lp * 32];
      const float* Cm = Bm + NST;
      float y = 0.0f;
#pragma unroll
      for (int n = 0; n < NST; ++n) {
        const float dA = __expf(dtv * Arow[n]);
        h[n] = dA * h[n] + u * Bm[n];
        y = fmaf(h[n], Cm[n], y);
      }
      const float z = xz[row * (2 * DMODEL) + DMODEL + c];
      xc[row * DMODEL + c] = (y + dskip * xcv) * silu_f(z);
    }
    __syncthreads();
  }
}

// ---------------------------------------------------------------------------
// Kernel 6: out[b,p,:] = scale/4 * sum_d ( yz_d[row_d(p)] @ out_w_d^T )
// Register-blocked: 4 N-tiles (64 output cols) per wave.
// grid = (4, BATCH*LSEQ/16), block = 1 wave
// ---------------------------------------------------------------------------
__global__ __launch_bounds__(32) void k_outproj(const float* __restrict__ yz,
                                                const float* __restrict__ out_w,
                                                const float* __restrict__ scale,
                                                float* __restrict__ out) {
  const int ntg   = blockIdx.x;                // 0..3, 64 cols each
  const int mtile = blockIdx.y;                // 0..255
  const int b  = mtile >> 6;
  const int l0 = (mtile & 63) << 4;
  const int t = threadIdx.x;
  const int mrow = t & 15, khalf = (t >> 4) << 1, col = t & 15, hi = t >> 4;
  const int n0 = ntg << 6;

  const int p  = l0 + mrow;
  const int hh = p >> 5, ww = p & 31;

  v8f a0 = {}, a1 = {}, a2 = {}, a3 = {};
  for (int dir = 0; dir < NDIR; ++dir) {
    int l;
    if (dir == 0)      l = p;
    else if (dir == 1) l = (LSEQ - 1) - p;
    else {
      const int lv = ww * HH + hh;             // inverse of the H/W transpose
      l = (dir == 2) ? lv : (LSEQ - 1) - lv;
    }
    const float* ar = yz + ((size_t)(dir * BATCH + b) * LSEQ + l) * DMODEL;
    const float* w0 = out_w + ((size_t)dir * DMODEL + n0 + col) * DMODEL;
    const float* w1 = w0 + (size_t)16 * DMODEL;
    const float* w2 = w0 + (size_t)32 * DMODEL;
    const float* w3 = w0 + (size_t)48 * DMODEL;
    for (int k0 = 0; k0 < DMODEL; k0 += 4) {
      const int ka = k0 + khalf;
      v2f a;  a.x = ar[ka]; a.y = ar[ka + 1];
      v2f b0; b0.x = w0[ka]; b0.y = w0[ka + 1];
      v2f b1; b1.x = w1[ka]; b1.y = w1[ka + 1];
      v2f b2; b2.x = w2[ka]; b2.y = w2[ka + 1];
      v2f b3; b3.x = w3[ka]; b3.y = w3[ka + 1];
      a0 = WMMA_F32(a, b0, a0);
      a1 = WMMA_F32(a, b1, a1);
      a2 = WMMA_F32(a, b2, a2);
      a3 = WMMA_F32(a, b3, a3);
    }
  }
  const float s = scale[0] * 0.25f;
#pragma unroll
  for (int r = 0; r < 8; ++r) {
    float* orow = out + ((size_t)b * LSEQ + l0 + hi * 8 + r) * DMODEL + n0 + col;
    orow[0]  = a0[r] * s;
    orow[16] = a1[r] * s;
    orow[32] = a2[r] * s;
    orow[48] = a3[r] * s;
  }
}

// ---------------------------------------------------------------------------
extern "C" void kernel_launch(void* const* d_in, const int* in_sizes, int n_in,
                              void* d_out, int out_size, void* d_ws, size_t ws_size,
                              hipStream_t stream) {
  const float* x      = (const float*)d_in[0];   // (B, D, H, W)
  const float* in_w   = (const float*)d_in[1];   // (4, 2D, D)
  const float* cw     = (const float*)d_in[2];   // (4, D, 4)
  const float* cb     = (const float*)d_in[3];   // (4, D)
  const float* xp_w   = (const float*)d_in[4];   // (4, 48, D)
  const float* dt_w   = (const float*)d_in[5];   // (4, D, 16)
  const float* dt_b   = (const float*)d_in[6];   // (4, D)
  const float* A_log  = (const float*)d_in[7];   // (4, D, 16)
  const float* Dp     = (const float*)d_in[8];   // (4, D)
  const float* out_w  = (const float*)d_in[9];   // (4, D, D)
  const float* scale  = (const float*)d_in[10];  // (1,)
  float* out = (float*)d_out;

  float* ws = (float*)d_ws;
  const size_t ROWS = (size_t)NDIR * BATCH * LSEQ;   // 16384
  float* xz  = ws;                                   // ROWS * 512
  float* xc  = xz  + ROWS * 2 * DMODEL;              // ROWS * 256 (becomes yz)
  float* dbl = xc  + ROWS * DMODEL;                  // ROWS * 48
  float* dt  = dbl + ROWS * XPO;                     // ROWS * 256

  // 1) in-projection (WMMA f32, 4 N-tiles per wave)
  k_inproj<<<dim3(2 * DMODEL / 64, ROWS / 16), 32, 0, stream>>>(x, in_w, xz);
  // 2) depthwise conv + SiLU
  {
    const size_t total = ROWS * DMODEL;
    k_conv<<<(unsigned)((total + 255) / 256), 256, 0, stream>>>(xz, cw, cb, xc);
  }
  // 3) x-projection (WMMA f32, all 3 N-tiles per wave)
  k_xproj<<<dim3(ROWS / 16), 32, 0, stream>>>(xc, xp_w, dbl);
  // 4) dt-projection + softplus (WMMA f32, 4 N-tiles per wave)
  k_dtproj<<<dim3(DMODEL / 64, ROWS / 16), 32, 0, stream>>>(dbl, dt_w, dt_b, dt);
  // 5) selective scan + gate (writes yz in place of xc)
  k_scan<<<NDIR * BATCH, 256, 0, stream>>>(dbl, dt, xz, A_log, Dp, xc);
  // 6) out-projection, 4-direction gather-accumulate (WMMA f32, 4 N-tiles per wave)
  k_outproj<<<dim3(DMODEL / 64, (size_t)BATCH * LSEQ / 16), 32, 0, stream>>>(
      xc, out_w, scale, out);
}